// TriggerLayer_22531398434885
// MI455X (gfx1250) — compile-verified
//
#include <hip/hip_runtime.h>
#include <cstdint>
#include <cstddef>

// Problem shape (fixed by the reference): B=256, H=512, WIDTH=512, WS=32.
#define TPB 256
#define IMG_H 512
#define IMG_W 512
#define WS 32

typedef __attribute__((ext_vector_type(4))) float v4f;

#ifndef __has_builtin
#define __has_builtin(x) 0
#endif

// ---- gfx1250 async LDS copy path (probe-confirmed to exist) ----------------
#if defined(__HIP_DEVICE_COMPILE__) &&                                         \
    __has_builtin(__builtin_amdgcn_global_load_async_to_lds_b128) &&           \
    __has_builtin(__builtin_amdgcn_global_store_async_from_lds_b128)
#define USE_ASYNC_LDS 1
#else
#define USE_ASYNC_LDS 0
#endif

#if defined(__HIP_DEVICE_COMPILE__)
__device__ __forceinline__ void wait_asynccnt0() {
#if __has_builtin(__builtin_amdgcn_s_wait_asynccnt)
  __builtin_amdgcn_s_wait_asynccnt(0);
#else
  asm volatile("s_wait_asynccnt 0x0" ::: "memory");
#endif
}
#endif

#if USE_ASYNC_LDS
// Builtin signature (from hipcc diagnostic): b128 variants take
// (v4i addrspace(1)*, v4i addrspace(3)*, imm offset, imm cpol).
typedef __attribute__((ext_vector_type(4))) int v4i;
typedef __attribute__((address_space(1))) v4i* gbl_v4i_p;
typedef __attribute__((address_space(3))) v4i* lds_v4i_p;
typedef __attribute__((address_space(1))) const v4f cgbl_v4f;
typedef __attribute__((address_space(3))) v4f lds_v4f;

__device__ __forceinline__ gbl_v4i_p as_global_v4i(const v4f* p) {
  // generic -> AS1 addrspacecast (same pointee), then reinterpret/const-strip.
  return (gbl_v4i_p)(cgbl_v4f*)p;
}
__device__ __forceinline__ lds_v4i_p as_lds_v4i(v4f* p) {
  // generic -> AS3 addrspacecast (same pointee), then reinterpret.
  return (lds_v4i_p)(lds_v4f*)p;
}
#endif

// Bulk streaming copy: images -> out, in 16B units. 256 threads = 8 wave32s.
__global__ void __launch_bounds__(TPB)
copy_kernel(const v4f* __restrict__ src, v4f* __restrict__ dst, long n4) {
#if USE_ASYNC_LDS
  // Stage 4 x 256 x 16B = 16KB through LDS via the CDNA5 async path.
  // Per-lane disjoint LDS slots; ASYNCcnt is per-wave, and each lane only
  // stores from the slot its own load filled, so s_wait_asynccnt 0 between
  // the load batch and store batch (and before buffer reuse) is sufficient.
  __shared__ v4f lbuf[4][TPB];
  const int  t      = threadIdx.x;
  const long stride = (long)gridDim.x * (4 * TPB);
  long base = (long)blockIdx.x * (4 * TPB) + t;
  for (; base < n4; base += stride) {
#pragma unroll
    for (int s = 0; s < 4; ++s) {
      long idx = base + (long)s * TPB;
      if (idx < n4) {
        __builtin_amdgcn_global_load_async_to_lds_b128(
            as_global_v4i(src + idx), as_lds_v4i(&lbuf[s][t]), 0, 0);
      }
    }
    wait_asynccnt0();  // loads into LDS complete before stores read it
#pragma unroll
    for (int s = 0; s < 4; ++s) {
      long idx = base + (long)s * TPB;
      if (idx < n4) {
        __builtin_amdgcn_global_store_async_from_lds_b128(
            as_global_v4i(dst + idx), as_lds_v4i(&lbuf[s][t]), 0, 0);
      }
    }
    wait_asynccnt0();  // stores drained before next iteration reuses lbuf
  }
#else
  // Fallback: 4-way unrolled non-temporal b128 copy + prefetch of next chunk.
  const long stride = (long)gridDim.x * (4 * TPB);
  long base = (long)blockIdx.x * (4 * TPB) + threadIdx.x;
  for (; base < n4; base += stride) {
    if (base + stride < n4) __builtin_prefetch(&src[base + stride], 0, 0);
    v4f v[4];
#pragma unroll
    for (int s = 0; s < 4; ++s) {
      long idx = base + (long)s * TPB;
      if (idx < n4) v[s] = __builtin_nontemporal_load(&src[idx]);
    }
#pragma unroll
    for (int s = 0; s < 4; ++s) {
      long idx = base + (long)s * TPB;
      if (idx < n4) __builtin_nontemporal_store(v[s], &dst[idx]);
    }
  }
#endif
}

// Paste W (32x32 f32) into each image at (p0, p1). One block per image,
// 256 threads; each thread covers 4 consecutive columns of one patch row.
// W reads are 16B-aligned float4; destination is only 4B-aligned (p1 is an
// arbitrary column), so the 4 output stores are scalar b32.
__global__ void __launch_bounds__(TPB)
patch_kernel(const float* __restrict__ W, const int* __restrict__ pos,
             float* __restrict__ out) {
  const int b  = blockIdx.x;
  const int p0 = pos[2 * b];      // uniform per block -> scalar load
  const int p1 = pos[2 * b + 1];  // uniform per block -> scalar load
  const int t  = threadIdx.x;
  const int r  = t >> 3;          // patch row   0..31
  const int c0 = (t & 7) * 4;     // patch col   0,4,...,28

  const v4f w = *(const v4f*)(W + r * WS + c0);
  float* dst = out + (size_t)b * IMG_H * IMG_W + (size_t)(p0 + r) * IMG_W
               + (size_t)p1 + c0;
  dst[0] = w.x;
  dst[1] = w.y;
  dst[2] = w.z;
  dst[3] = w.w;
}

extern "C" void kernel_launch(void* const* d_in, const int* in_sizes, int n_in,
                              void* d_out, int out_size, void* d_ws,
                              size_t ws_size, hipStream_t stream) {
  (void)n_in; (void)out_size; (void)d_ws; (void)ws_size;

  const float* images = (const float*)d_in[0];  // (256, 512, 512) f32
  const int*   pos    = (const int*)d_in[1];    // (256, 2) i32
  const float* W      = (const float*)d_in[2];  // (32, 32) f32
  float*       out    = (float*)d_out;          // (256, 512, 512) f32

  const long n  = (long)in_sizes[0];            // 67,108,864 floats
  const long n4 = n / 4;                        // 16,777,216 x 16B
  const int  nb = in_sizes[1] / 2;              // 256 images

  // 8192 blocks x 8 waves = 65536 wave32s in flight: enough outstanding
  // b128 requests to saturate 23.3 TB/s HBM on a pure streaming copy.
  copy_kernel<<<8192, TPB, 0, stream>>>((const v4f*)images, (v4f*)out, n4);
  // Stream-ordered after the bulk copy (WAW on the patch region).
  patch_kernel<<<nb, TPB, 0, stream>>>(W, pos, out);
}